// PGExplainer_55078660604627
// MI455X (gfx1250) — compile-verified
//
#include <hip/hip_runtime.h>

typedef float v2f __attribute__((ext_vector_type(2)));
typedef float v8f __attribute__((ext_vector_type(8)));

#define DIM   128   // D
#define HID   128   // H
#define MCAND 512   // candidates per query

// Swizzled index into the transposed W1-half staged in LDS.
// Column-major (c = output column 0..127, k = K within half 0..127) so the
// two K values of a B fragment are adjacent (one ds_load_b64); the XOR term
// spreads the 16 columns of a half-wave across banks (c*128 alone aliases
// every column onto the same bank pair).
__device__ __forceinline__ int lds_idx(int c, int k) {
    return c * 128 + (k ^ ((c & 7) << 4));
}

// ---------------------------------------------------------------------------
// Kernel 0: Hq[b, :] = q_b @ W1[0:128, :] + b1     (tiny: 1024x128x128)
// One wave per 16 queries; WMMA f32 16x16x4.
// ---------------------------------------------------------------------------
__global__ __launch_bounds__(256)
void pg_hq_kernel(const float* __restrict__ embed,
                  const float* __restrict__ W1,
                  const float* __restrict__ b1,
                  const int*   __restrict__ query_idx,
                  float*       __restrict__ Hq)
{
    const int lane = threadIdx.x & 31;
    const int wave = threadIdx.x >> 5;
    const int row0 = (blockIdx.x * 8 + wave) * 16;   // 16 queries per tile
    const int col  = lane & 15;
    const int koff = (lane >> 4) << 1;

    const float* arow = embed + (size_t)query_idx[row0 + col] * DIM;

    v8f acc[8] = {};
    #pragma unroll 4
    for (int kk = 0; kk < 32; ++kk) {
        const int k0 = kk * 4 + koff;
        v2f a;
        a.x = arow[k0];
        a.y = arow[k0 + 1];
        #pragma unroll
        for (int n = 0; n < 8; ++n) {
            v2f bf;
            bf.x = W1[(size_t)k0       * HID + n * 16 + col];
            bf.y = W1[(size_t)(k0 + 1) * HID + n * 16 + col];
            acc[n] = __builtin_amdgcn_wmma_f32_16x16x4_f32(
                false, a, false, bf, (short)0, acc[n], false, false);
        }
    }

    const int hi = (lane >> 4) << 3;
    #pragma unroll
    for (int n = 0; n < 8; ++n) {
        const float b1v = b1[n * 16 + col];
        #pragma unroll
        for (int r = 0; r < 8; ++r)
            Hq[(size_t)(row0 + r + hi) * HID + n * 16 + col] = acc[n][r] + b1v;
    }
}

// ---------------------------------------------------------------------------
// Kernel 1: candidate-half GEMM + fused epilogue.
// Each wave owns TWO adjacent 16-edge tiles (same query -> shared Hq and
// shared B fragments). W1 rows 128..255 staged once per block in 64KB LDS.
// scores[edge] = relu(c_e @ W1_bot + Hq[b]) @ W2 + b2
// ---------------------------------------------------------------------------
__global__ __launch_bounds__(256)
void pg_score_kernel(const float* __restrict__ embed,
                     const float* __restrict__ W1,
                     const float* __restrict__ W2,
                     const float* __restrict__ b2,
                     const int*   __restrict__ cand_idx,
                     const float* __restrict__ Hq,
                     float*       __restrict__ scores)
{
    __shared__ float ldsW[128 * 128];   // 64KB: W1[128:256, :] transposed+swizzled

    for (int i = threadIdx.x; i < 128 * 128; i += 256) {
        const int k = i >> 7;       // K within bottom half
        const int c = i & 127;      // output column
        ldsW[lds_idx(c, k)] = W1[(size_t)(k + DIM) * HID + c];
    }
    __syncthreads();

    const int lane  = threadIdx.x & 31;
    const int wave  = threadIdx.x >> 5;
    const int tile0 = (blockIdx.x * 8 + wave) * 2;   // even start: pair shares bq
    const int edgeA = tile0 * 16;
    const int edgeB = edgeA + 16;
    const int bq    = edgeA >> 9;
    const int col   = lane & 15;
    const int koff  = (lane >> 4) << 1;

    const float* crowA = embed + (size_t)cand_idx[edgeA + col] * DIM;
    const float* crowB = embed + (size_t)cand_idx[edgeB + col] * DIM;

    v8f acc0[8] = {};
    v8f acc1[8] = {};

    #pragma unroll 4
    for (int kk = 0; kk < 32; ++kk) {
        const int k0 = kk * 4 + koff;
        v2f a0, a1;
        a0.x = crowA[k0];
        a0.y = crowA[k0 + 1];
        a1.x = crowB[k0];
        a1.y = crowB[k0 + 1];
        #pragma unroll
        for (int n = 0; n < 8; ++n) {
            const v2f bf = *(const v2f*)&ldsW[lds_idx(n * 16 + col, k0)];
            acc0[n] = __builtin_amdgcn_wmma_f32_16x16x4_f32(
                false, a0, false, bf, (short)0, acc0[n], false, false);
            acc1[n] = __builtin_amdgcn_wmma_f32_16x16x4_f32(
                false, a1, false, bf, (short)0, acc1[n], false, false);
        }
    }

    // fused epilogue: h = relu(acc + Hq), score = h . W2 + b2
    const float* hqrow = Hq + (size_t)bq * HID;
    float hqv[8], w2v[8];
    #pragma unroll
    for (int n = 0; n < 8; ++n) {
        hqv[n] = hqrow[n * 16 + col];
        w2v[n] = W2[n * 16 + col];
    }
    const float b2s = b2[0];
    const int   hi  = (lane >> 4) << 3;

    #pragma unroll
    for (int r = 0; r < 8; ++r) {
        float p0 = 0.0f, p1 = 0.0f;
        #pragma unroll
        for (int n = 0; n < 8; ++n) {
            float h0 = acc0[n][r] + hqv[n];
            float h1 = acc1[n][r] + hqv[n];
            h0 = h0 > 0.0f ? h0 : 0.0f;
            h1 = h1 > 0.0f ? h1 : 0.0f;
            p0 += h0 * w2v[n];
            p1 += h1 * w2v[n];
        }
        p0 += __shfl_xor(p0, 1);  p1 += __shfl_xor(p1, 1);
        p0 += __shfl_xor(p0, 2);  p1 += __shfl_xor(p1, 2);
        p0 += __shfl_xor(p0, 4);  p1 += __shfl_xor(p1, 4);
        p0 += __shfl_xor(p0, 8);  p1 += __shfl_xor(p1, 8);
        if (col == 0) {
            scores[edgeA + r + hi] = p0 + b2s;
            scores[edgeB + r + hi] = p1 + b2s;
        }
    }
}

// ---------------------------------------------------------------------------
// Kernel 2: bulk copy node_feat -> masked_feat output (float4, grid-stride)
// ---------------------------------------------------------------------------
__global__ __launch_bounds__(256)
void pg_copy_kernel(const float4* __restrict__ src,
                    float4*       __restrict__ dst,
                    int n4)
{
    int i = blockIdx.x * blockDim.x + threadIdx.x;
    const int stride = gridDim.x * blockDim.x;
    for (; i < n4; i += stride) dst[i] = src[i];
}

// ---------------------------------------------------------------------------
// Kernel 3: per-query top-k scan (strict '>' keeps earliest index, matching
// lax.top_k tie-breaking), then zero the selected nodes' feature rows.
// ---------------------------------------------------------------------------
__global__ __launch_bounds__(256)
void pg_topk_zero_kernel(const float* __restrict__ scores,
                         const int*   __restrict__ cand_idx,
                         const int*   __restrict__ topk_ptr,
                         float*       __restrict__ feat_out,
                         int B)
{
    const int bq = blockIdx.x * blockDim.x + threadIdx.x;
    if (bq >= B) return;

    int k = *topk_ptr;
    if (k > 16) k = 16;
    if (k < 0)  k = 0;

    const float* s = scores + (size_t)bq * MCAND;
    int chosen[16];
    for (int j = 0; j < k; ++j) {
        float best = -__builtin_huge_valf();
        int   bi   = 0;
        for (int m = 0; m < MCAND; ++m) {
            bool skip = false;
            for (int t = 0; t < j; ++t)
                if (chosen[t] == m) skip = true;
            const float v = s[m];
            if (!skip && v > best) { best = v; bi = m; }
        }
        chosen[j] = bi;
    }

    for (int j = 0; j < k; ++j) {
        const int node = cand_idx[(size_t)bq * MCAND + chosen[j]];
        float4* row = (float4*)(feat_out + (size_t)node * DIM);
        #pragma unroll
        for (int i = 0; i < DIM / 4; ++i)
            row[i] = make_float4(0.f, 0.f, 0.f, 0.f);
    }
}

// ---------------------------------------------------------------------------
extern "C" void kernel_launch(void* const* d_in, const int* in_sizes, int n_in,
                              void* d_out, int out_size, void* d_ws, size_t ws_size,
                              hipStream_t stream)
{
    const float* embed     = (const float*)d_in[0];
    const float* node_feat = (const float*)d_in[1];
    const float* W1        = (const float*)d_in[2];
    const float* b1        = (const float*)d_in[3];
    const float* W2        = (const float*)d_in[4];
    const float* b2        = (const float*)d_in[5];
    const int*   query_idx = (const int*)d_in[6];
    const int*   cand_idx  = (const int*)d_in[7];
    const int*   topk_ptr  = (const int*)d_in[8];

    const int B     = in_sizes[6];            // 1024
    const int edges = in_sizes[7];            // B * 512
    const int featN = in_sizes[1];            // N * 128

    float* scores   = (float*)d_out;          // [B, M]
    float* feat_out = (float*)d_out + edges;  // [N, D]
    float* Hq       = (float*)d_ws;           // [B, H] scratch (512KB)

    // 0) per-query hidden precompute (hoists the shared query-half GEMM)
    pg_hq_kernel<<<B / (16 * 8), 256, 0, stream>>>(
        embed, W1, b1, query_idx, Hq);

    // 1) candidate-half WMMA GEMM + fused score epilogue
    //    2 tiles/wave * 8 waves = 16 edge-tiles (256 edges) ... per block:
    //    8 waves * 2 tiles * 16 edges = 256 edges -> edges/256 blocks
    pg_score_kernel<<<edges / 256, 256, 0, stream>>>(
        embed, W1, W2, b2, cand_idx, Hq, scores);

    // 2) copy node_feat into output (fully rewrites the poisoned region)
    {
        const int n4 = featN / 4;
        int blocks = (n4 + 255) / 256;
        if (blocks > 16384) blocks = 16384;
        pg_copy_kernel<<<blocks, 256, 0, stream>>>(
            (const float4*)node_feat, (float4*)feat_out, n4);
    }

    // 3) top-k selection + hard mask (ordered after the copy on `stream`)
    pg_topk_zero_kernel<<<(B + 255) / 256, 256, 0, stream>>>(
        scores, cand_idx, topk_ptr, feat_out, B);
}